// GNN_20882130993433
// MI455X (gfx1250) — compile-verified
//
#include <hip/hip_runtime.h>

// GraphSAGE 2-layer + mean-pool + FC for MI455X (gfx1250, wave32).
// Aggregation via wave-per-edge float atomics (L2-resident); GEMMs via
// V_WMMA_F32_16X16X4_F32 with transposed weights (contiguous b64 B loads)
// and 4 row-tiles per wave to amortize B traffic 4 WMMAs per B fragment.

typedef float v2f __attribute__((ext_vector_type(2)));
typedef float v8f __attribute__((ext_vector_type(8)));

#define D_H        128
#define D_OUT      10
#define NUM_GRAPHS 512
#define ROWS_PER_BLOCK 64          // 4 row tiles of 16

// ---------------------------------------------------------------- zero fill
__global__ void gnn_zero_f32(float* __restrict__ p, int n) {
  int i = blockIdx.x * blockDim.x + threadIdx.x;
  if (i < n) p[i] = 0.0f;
}

// ------------------------------------------------ weight transpose helpers
// w: [128 x 128] row-major (k x n)  ->  wt: [128 x 128] (n x k)
__global__ void gnn_transpose128(const float* __restrict__ w,
                                 float* __restrict__ wt) {
  int idx = blockIdx.x * blockDim.x + threadIdx.x;   // 16384
  int n = idx >> 7, k = idx & 127;
  wt[idx] = w[k * D_H + n];
}

// fc_w: [128 x 10] -> fwt: [16 x 128], rows 10..15 zero-padded
__global__ void gnn_transpose_fc(const float* __restrict__ fw,
                                 float* __restrict__ fwt) {
  int idx = blockIdx.x * blockDim.x + threadIdx.x;   // 2048
  int n = idx >> 7, k = idx & 127;
  fwt[idx] = (n < D_OUT) ? fw[k * D_OUT + n] : 0.0f;
}

// ------------------------------------------------- edge scatter (mean aggr)
// One wave32 per edge: 32 lanes x float4 = 128 floats per node row.
__global__ void gnn_scatter_edges(const float* __restrict__ xin,
                                  const long long* __restrict__ src,
                                  const long long* __restrict__ tgt,
                                  float* __restrict__ agg,
                                  float* __restrict__ deg,
                                  int E, int with_deg) {
  int wid  = (blockIdx.x * blockDim.x + threadIdx.x) >> 5;  // edge id
  int lane = threadIdx.x & 31;
  if (wid >= E) return;
  long long s = src[wid];
  long long t = tgt[wid];
  float4 v = ((const float4*)(xin + (size_t)s * D_H))[lane];
  float* ar = agg + (size_t)t * D_H + lane * 4;
  atomicAdd(ar + 0, v.x);
  atomicAdd(ar + 1, v.y);
  atomicAdd(ar + 2, v.z);
  atomicAdd(ar + 3, v.w);
  if (with_deg && lane == 0) atomicAdd(deg + t, 1.0f);
}

// --------------------------------------------- fused SAGE combine via WMMA
// out[m,:] = (agg[m,:]/max(deg,1)) @ Wn + b + xin[m,:] @ Wr   (+ relu)
// Block = 256 threads = 8 waves; 64 rows (4 row tiles) per block; wave w
// owns the 16-column tile at 16*w. A staged in LDS; B from transposed
// weights, one b64 load per matrix per k-step, reused by 4 row tiles.
__global__ __launch_bounds__(256)
void gnn_sage_combine(const float* __restrict__ xin,
                      const float* __restrict__ agg,
                      const float* __restrict__ deg,
                      const float* __restrict__ wnt,   // transposed [n][k]
                      const float* __restrict__ bias,
                      const float* __restrict__ wrt,   // transposed [n][k]
                      float* __restrict__ out,
                      int N, int do_relu) {
  __shared__ float sN[ROWS_PER_BLOCK * D_H];   // mean-aggregated neighbors
  __shared__ float sR[ROWS_PER_BLOCK * D_H];   // root (self) features

  const int m0 = blockIdx.x * ROWS_PER_BLOCK;
  const int t  = threadIdx.x;

  // ---- stage A tiles: 4 threads per row, 32-float chunks, scale by 1/deg
  {
    const int srow = t >> 2;               // 0..63
    const int scol = (t & 3) * 32;         // 0,32,64,96
    int gr = m0 + srow; if (gr > N - 1) gr = N - 1;     // clamped read
    const float ds = 1.0f / fmaxf(deg[gr], 1.0f);
    const float4* aN = (const float4*)(agg + (size_t)gr * D_H + scol);
    const float4* aR = (const float4*)(xin + (size_t)gr * D_H + scol);
    float4* dN = (float4*)(sN + srow * D_H + scol);
    float4* dR = (float4*)(sR + srow * D_H + scol);
    #pragma unroll
    for (int i = 0; i < 8; ++i) {
      float4 v = aN[i];
      v.x *= ds; v.y *= ds; v.z *= ds; v.w *= ds;
      dN[i] = v;
      dR[i] = aR[i];
    }
  }
  __syncthreads();

  const int wave  = t >> 5;                // 0..7 -> column tile
  const int lane  = t & 31;
  const int n0c   = wave * 16;
  const int mrow  = lane & 15;             // A row within a 16-row tile
  const int khalf = (lane >> 4) * 2;       // 0 or 2
  const int ncol  = n0c + (lane & 15);     // B/D column

  const float* bn_base = wnt + (size_t)ncol * D_H;
  const float* br_base = wrt + (size_t)ncol * D_H;

  v8f acc[4] = {};
  #pragma unroll 4
  for (int k = 0; k < D_H; k += 4) {
    const int kb = k + khalf;
    const v2f b_n = *(const v2f*)(bn_base + kb);
    const v2f b_r = *(const v2f*)(br_base + kb);
    #pragma unroll
    for (int rt = 0; rt < 4; ++rt) {
      const int arow = (rt * 16 + mrow) * D_H + kb;
      acc[rt] = __builtin_amdgcn_wmma_f32_16x16x4_f32(
          false, *(const v2f*)(sN + arow), false, b_n, (short)0, acc[rt],
          false, false);
      acc[rt] = __builtin_amdgcn_wmma_f32_16x16x4_f32(
          false, *(const v2f*)(sR + arow), false, b_r, (short)0, acc[rt],
          false, false);
    }
  }

  const float bv = bias[ncol];
  #pragma unroll
  for (int rt = 0; rt < 4; ++rt) {
    const int rbase = m0 + rt * 16 + (lane >> 4) * 8;
    #pragma unroll
    for (int r = 0; r < 8; ++r) {
      float v = acc[rt][r] + bv;
      if (do_relu) v = fmaxf(v, 0.0f);
      const int row = rbase + r;
      if (row < N) out[(size_t)row * D_H + ncol] = v;   // guard after WMMA
    }
  }
}

// --------------------------------------------------------- graph mean pool
__global__ void gnn_pool_scatter(const float* __restrict__ h,
                                 const long long* __restrict__ batch,
                                 float* __restrict__ pooled,
                                 float* __restrict__ cnt, int N) {
  int wid  = (blockIdx.x * blockDim.x + threadIdx.x) >> 5;  // node id
  int lane = threadIdx.x & 31;
  if (wid >= N) return;
  long long g = batch[wid];
  float4 v = ((const float4*)(h + (size_t)wid * D_H))[lane];
  float* pr = pooled + (size_t)g * D_H + lane * 4;
  atomicAdd(pr + 0, v.x);
  atomicAdd(pr + 1, v.y);
  atomicAdd(pr + 2, v.z);
  atomicAdd(pr + 3, v.w);
  if (lane == 0) atomicAdd(cnt + g, 1.0f);
}

// ---------------------------------------- final FC: (pooled/cnt)@W + b
// One wave per 16-row tile. fwt is transposed + zero-padded to 16 columns,
// so no masking is needed and EXEC is all-1s through the WMMA chain.
__global__ void gnn_pooled_fc(const float* __restrict__ pooled,
                              const float* __restrict__ cnt,
                              const float* __restrict__ fwt,  // [16 x 128]
                              const float* __restrict__ fb,
                              float* __restrict__ out) {
  const int m0    = blockIdx.x * 16;
  const int lane  = threadIdx.x;           // blockDim.x == 32
  const int mrow  = m0 + (lane & 15);
  const int khalf = (lane >> 4) * 2;
  const int n     = lane & 15;
  const float sc  = 1.0f / fmaxf(cnt[mrow], 1.0f);
  const float* arow = pooled + (size_t)mrow * D_H;
  const float* brow = fwt + (size_t)n * D_H;

  v8f c = {};
  #pragma unroll
  for (int k = 0; k < D_H; k += 4) {
    const int kb = k + khalf;
    v2f a = *(const v2f*)(arow + kb);
    a.x *= sc; a.y *= sc;
    const v2f b = *(const v2f*)(brow + kb);
    c = __builtin_amdgcn_wmma_f32_16x16x4_f32(false, a, false, b,
                                              (short)0, c, false, false);
  }

  if (n < D_OUT) {
    const float bv = fb[n];
    const int rb = m0 + (lane >> 4) * 8;
    #pragma unroll
    for (int r = 0; r < 8; ++r)
      out[(size_t)(rb + r) * D_OUT + n] = c[r] + bv;
  }
}

// --------------------------------------------------------------- dispatch
extern "C" void kernel_launch(void* const* d_in, const int* in_sizes, int n_in,
                              void* d_out, int out_size, void* d_ws, size_t ws_size,
                              hipStream_t stream) {
  const float*     x     = (const float*)d_in[0];
  const long long* ei    = (const long long*)d_in[1];   // int64 [2,E]
  const long long* batch = (const long long*)d_in[2];   // int64 [N]
  const float*     w1n   = (const float*)d_in[3];
  const float*     b1    = (const float*)d_in[4];
  const float*     w1r   = (const float*)d_in[5];
  const float*     w2n   = (const float*)d_in[6];
  const float*     b2    = (const float*)d_in[7];
  const float*     w2r   = (const float*)d_in[8];
  const float*     fcw   = (const float*)d_in[9];
  const float*     fcb   = (const float*)d_in[10];
  float* out = (float*)d_out;

  const int N = in_sizes[0] / D_H;
  const int E = in_sizes[1] / 2;
  const long long* src = ei;
  const long long* tgt = ei + E;

  // Workspace (floats): agg[N*128] h1[N*128] deg[N] pooled[512*128] cnt[512]
  //                     wnt[16384] wrt[16384] fwt[2048]
  float* ws     = (float*)d_ws;
  float* agg    = ws;
  float* h1     = agg + (size_t)N * D_H;
  float* deg    = h1  + (size_t)N * D_H;
  float* pooled = deg + N;
  float* cnt    = pooled + (size_t)NUM_GRAPHS * D_H;
  float* wnt    = cnt + NUM_GRAPHS;
  float* wrt    = wnt + D_H * D_H;
  float* fwt    = wrt + D_H * D_H;

  const int nd       = N * D_H;
  const int combBlks = (N + ROWS_PER_BLOCK - 1) / ROWS_PER_BLOCK;

  // ---- layer 1
  gnn_zero_f32<<<(nd + 255) / 256, 256, 0, stream>>>(agg, nd);
  gnn_zero_f32<<<(N + 255) / 256, 256, 0, stream>>>(deg, N);
  gnn_transpose128<<<(D_H * D_H) / 256, 256, 0, stream>>>(w1n, wnt);
  gnn_transpose128<<<(D_H * D_H) / 256, 256, 0, stream>>>(w1r, wrt);
  gnn_scatter_edges<<<(E + 7) / 8, 256, 0, stream>>>(x, src, tgt, agg, deg, E, 1);
  gnn_sage_combine<<<combBlks, 256, 0, stream>>>(x, agg, deg, wnt, b1, wrt, h1, N, 1);

  // ---- layer 2 (deg reused; combine writes in-place over agg)
  gnn_zero_f32<<<(nd + 255) / 256, 256, 0, stream>>>(agg, nd);
  gnn_transpose128<<<(D_H * D_H) / 256, 256, 0, stream>>>(w2n, wnt);
  gnn_transpose128<<<(D_H * D_H) / 256, 256, 0, stream>>>(w2r, wrt);
  gnn_scatter_edges<<<(E + 7) / 8, 256, 0, stream>>>(h1, src, tgt, agg, deg, E, 0);
  gnn_sage_combine<<<combBlks, 256, 0, stream>>>(h1, agg, deg, wnt, b2, wrt, agg, N, 0);

  // ---- global mean pool + FC head
  gnn_zero_f32<<<(NUM_GRAPHS * D_H + 255) / 256, 256, 0, stream>>>(pooled, NUM_GRAPHS * D_H);
  gnn_zero_f32<<<(NUM_GRAPHS + 255) / 256, 256, 0, stream>>>(cnt, NUM_GRAPHS);
  gnn_transpose_fc<<<(16 * D_H) / 256, 256, 0, stream>>>(fcw, fwt);
  gnn_pool_scatter<<<(N + 7) / 8, 256, 0, stream>>>(agg, batch, pooled, cnt, N);
  gnn_pooled_fc<<<NUM_GRAPHS / 16, 32, 0, stream>>>(pooled, cnt, fwt, fcb, out);
}